// MyGCNConv_35536559407736
// MI455X (gfx1250) — compile-verified
//
#include <hip/hip_runtime.h>

// GCNConv for MI455X (gfx1250):
//   Phase 1: h = x @ W via V_WMMA_F32_16X16X4_F32 (fp32 matrix pipe, exact dtype)
//   Phase 2: out[i] = (1/deg_i) * sum_{e in row i} h[idx[e]] + bias
// h (51.2 MB) lives in d_ws and stays resident in the 192 MB L2 between kernels,
// so the ~819 MB of random edge-gather traffic is served from L2, not HBM.

typedef float v2f __attribute__((ext_vector_type(2)));
typedef float v8f __attribute__((ext_vector_type(8)));

#define DIN  128
#define DOUT 128

// W staged in LDS with k-pair interleaving:
//   Wl[(k>>1)*PAIR_STRIDE + c*2 + (k&1)] = W[k][c]
// so a B fragment (W[k'][c], W[k'+1][c]) is one contiguous 8B ds_load_b64
// directly into the even-aligned VGPR pair WMMA wants (no v_mov shuffles).
// PAIR_STRIDE = 288: +1 pair-row => +288 elements => +32 banks, so lane
// group 1 (k'+2) uses the opposite 32 LDS banks -> conflict-free b64 loads.
#define PAIR_STRIDE 288
#define WL_FLOATS   (64 * PAIR_STRIDE)  // 72 KB

// ---------------------------------------------------------------------------
// Phase 1: dense GEMM h = x @ W using wave32 WMMA f32 16x16x4.
// Block = 256 threads = 8 waves; block covers 128 rows x 128 cols.
// Each wave: one 16-row strip, 8 n-tile v8f accumulators; one v2f A-fragment
// load feeds 8 WMMAs per k-step.
// ---------------------------------------------------------------------------
__global__ __launch_bounds__(256) void gcn_gemm_wmma(
    const float* __restrict__ x, const float* __restrict__ W,
    float* __restrict__ h, int n_rows) {
  __shared__ float Wl[WL_FLOATS];

  const int tid = threadIdx.x;
  // Cooperative staging with pair-interleave swizzle (16384 floats total).
  for (int i = tid * 4; i < DIN * DOUT; i += 256 * 4) {
    const int r = i >> 7;          // row k
    const int c = i & 127;         // col start (multiple of 4, same row)
    const float4 w = *(const float4*)(&W[i]);
    const int base = (r >> 1) * PAIR_STRIDE + (r & 1);
    Wl[base + (c + 0) * 2] = w.x;
    Wl[base + (c + 1) * 2] = w.y;
    Wl[base + (c + 2) * 2] = w.z;
    Wl[base + (c + 3) * 2] = w.w;
  }
  __syncthreads();

  const int wave = tid >> 5;
  const int lane = tid & 31;
  const int g    = lane >> 4;   // lane group (0: lanes 0-15, 1: lanes 16-31)
  const int ln   = lane & 15;

  const int rowBase = blockIdx.x * 128 + wave * 16;  // M-tile origin
  // A-fragment row (clamp only matters in the tail block; stores masked there).
  int arow = rowBase + ln;
  if (arow >= n_rows) arow = n_rows - 1;
  const float* xrow = x + (size_t)arow * DIN;

  v8f acc[8];
#pragma unroll
  for (int t = 0; t < 8; ++t) acc[t] = (v8f)0.0f;

  // ISA layout for 32-bit A (16x4) and B (4x16): K = vgpr_idx + 2*laneGroup.
  for (int k = 0; k < DIN; k += 4) {
    const v2f a = *(const v2f*)(xrow + k + 2 * g);           // global b64
    const int pbase = ((k + 2 * g) >> 1) * PAIR_STRIDE + ln * 2;
#pragma unroll
    for (int t = 0; t < 8; ++t) {
      const v2f b = *(const v2f*)(&Wl[pbase + t * 32]);      // ds_load_b64
      // 8 args: (neg_a, A, neg_b, B, c_mod, C, reuse_a, reuse_b)
      acc[t] = __builtin_amdgcn_wmma_f32_16x16x4_f32(
          false, a, false, b, (short)0, acc[t], false, false);
    }
  }

  // C/D layout: VGPR v, lane (g*16+ln) holds (row = rowBase + v + 8g, col = 16t + ln).
  if (rowBase + 16 <= n_rows) {
    // Fast path: whole wave-strip in range, unconditional stores.
#pragma unroll
    for (int t = 0; t < 8; ++t) {
#pragma unroll
      for (int v = 0; v < 8; ++v) {
        const int row = rowBase + v + 8 * g;
        h[(size_t)row * DOUT + t * 16 + ln] = acc[t][v];
      }
    }
  } else {
    // Tail block only (n_rows % 128 != 0): per-row masking.
#pragma unroll
    for (int t = 0; t < 8; ++t) {
#pragma unroll
      for (int v = 0; v < 8; ++v) {
        const int row = rowBase + v + 8 * g;
        if (row < n_rows) h[(size_t)row * DOUT + t * 16 + ln] = acc[t][v];
      }
    }
  }
}

// ---------------------------------------------------------------------------
// Phase 2: CSR mean-aggregate + bias. One wave32 per destination node.
// node is forced wave-uniform via readfirstlane so ptr/idx traffic goes down
// the SCALAR memory path (s_load, KMcnt, batched s_load_b128), leaving the
// vector pipe exclusively for the h gathers. Lane l owns output columns
// [4l, 4l+4): each gathered source row is one coalesced 16B load per lane
// (global_load_b128). Edge loop unrolled x4 with independent partial sums
// for memory-level parallelism against L2 gather latency.
// ---------------------------------------------------------------------------
__global__ __launch_bounds__(256) void gcn_aggregate(
    const float* __restrict__ h, const int* __restrict__ ptr,
    const int* __restrict__ idx, const float* __restrict__ bias,
    float* __restrict__ out, int n_nodes) {
  const int lane = threadIdx.x & 31;
  // Wave-uniform node id (provably uniform -> scalar loads for ptr/idx).
  const int node = __builtin_amdgcn_readfirstlane(
      (int)(blockIdx.x * 8 + (threadIdx.x >> 5)));
  if (node >= n_nodes) return;

  const int p0  = ptr[node];
  const int p1  = ptr[node + 1];
  const int deg = p1 - p0;
  const int c   = lane * 4;

  float4 a0 = make_float4(0.f, 0.f, 0.f, 0.f);
  float4 a1 = a0, a2 = a0, a3 = a0;

  int e = p0;
  for (; e + 4 <= p1; e += 4) {
    const int s0 = idx[e + 0];
    const int s1 = idx[e + 1];
    const int s2 = idx[e + 2];
    const int s3 = idx[e + 3];
    const float4 v0 = *(const float4*)(h + (size_t)s0 * DOUT + c);
    const float4 v1 = *(const float4*)(h + (size_t)s1 * DOUT + c);
    const float4 v2 = *(const float4*)(h + (size_t)s2 * DOUT + c);
    const float4 v3 = *(const float4*)(h + (size_t)s3 * DOUT + c);
    a0.x += v0.x; a0.y += v0.y; a0.z += v0.z; a0.w += v0.w;
    a1.x += v1.x; a1.y += v1.y; a1.z += v1.z; a1.w += v1.w;
    a2.x += v2.x; a2.y += v2.y; a2.z += v2.z; a2.w += v2.w;
    a3.x += v3.x; a3.y += v3.y; a3.z += v3.z; a3.w += v3.w;
  }
  for (; e < p1; ++e) {
    const int s = idx[e];
    const float4 v = *(const float4*)(h + (size_t)s * DOUT + c);
    a0.x += v.x; a0.y += v.y; a0.z += v.z; a0.w += v.w;
  }

  float4 acc;
  acc.x = (a0.x + a1.x) + (a2.x + a3.x);
  acc.y = (a0.y + a1.y) + (a2.y + a3.y);
  acc.z = (a0.z + a1.z) + (a2.z + a3.z);
  acc.w = (a0.w + a1.w) + (a2.w + a3.w);

  const float inv = (deg > 0) ? (1.0f / (float)deg) : 0.0f;  // gcn_norm, inf->0
  const float4 b = *(const float4*)(bias + c);
  float4 r;
  r.x = acc.x * inv + b.x;
  r.y = acc.y * inv + b.y;
  r.z = acc.z * inv + b.z;
  r.w = acc.w * inv + b.w;
  *(float4*)(out + (size_t)node * DOUT + c) = r;
}

// ---------------------------------------------------------------------------
// Launch. Inputs (setup_inputs order): x, W, bias, ptr, idx, num_node.
// ---------------------------------------------------------------------------
extern "C" void kernel_launch(void* const* d_in, const int* in_sizes, int n_in,
                              void* d_out, int out_size, void* d_ws, size_t ws_size,
                              hipStream_t stream) {
  const float* x    = (const float*)d_in[0];
  const float* W    = (const float*)d_in[1];
  const float* bias = (const float*)d_in[2];
  const int*   ptr  = (const int*)d_in[3];
  const int*   idx  = (const int*)d_in[4];

  const int n_nodes = in_sizes[0] / DIN;  // 100000
  float* h = (float*)d_ws;                // n_nodes * DOUT floats (51.2 MB)

  const int gemm_blocks = (n_nodes + 127) / 128;
  gcn_gemm_wmma<<<gemm_blocks, 256, 0, stream>>>(x, W, h, n_nodes);

  const int agg_blocks = (n_nodes + 7) / 8;
  gcn_aggregate<<<agg_blocks, 256, 0, stream>>>(h, ptr, idx, bias,
                                                (float*)d_out, n_nodes);
}